// QCNN_40707700032198
// MI455X (gfx1250) — compile-verified
//
#include <hip/hip_runtime.h>
#include <math.h>

typedef __attribute__((ext_vector_type(2))) float v2f;
typedef __attribute__((ext_vector_type(8))) float v8f;

#define N_QUBITS 8
#define INPUT_DIM 3072
#define NUM_CLASSES 10
#define DIM 256          // 2^8 statevector
#define BK 128           // K block for feats GEMM (double-buffered)
#define NB (INPUT_DIM / BK)          // 24 K-blocks
#define XS_STRIDE 132    // padded LDS row stride (floats): conflict-free A-frag b64 loads
#define TILE_F (16 * XS_STRIDE)      // floats per buffer
#define PS 260           // padded psi stride

// ---------- complex helpers ----------
struct c32 { float x, y; };
__device__ __forceinline__ c32 cmul(c32 a, c32 b){ return c32{a.x*b.x - a.y*b.y, a.x*b.y + a.y*b.x}; }
__device__ __forceinline__ c32 cadd(c32 a, c32 b){ return c32{a.x+b.x, a.y+b.y}; }

// ---------- WMMA wrapper: D = A(16x4,f32) * B(4x16,f32) + C ----------
__device__ __forceinline__ v8f wmma_f32_16x16x4(v2f a, v2f b, v8f c) {
  return __builtin_amdgcn_wmma_f32_16x16x4_f32(false, a, false, b, (short)0, c, false, false);
}

// ---------- CDNA5 async global->LDS DMA (ASYNCcnt tracked) ----------
__device__ __forceinline__ void async_ld_b128(const float* g, const float* lds) {
  unsigned l = (unsigned)(uintptr_t)lds;   // LDS address = low 32 bits of flat pointer
  asm volatile("global_load_async_to_lds_b128 %0, %1, off"
               :: "v"(l), "v"(g) : "memory");
}
__device__ __forceinline__ void wait_async0() {
  asm volatile("s_wait_asynccnt 0x0" ::: "memory");
}

// =====================================================================
// Kernel 0: pack fc_w[8][3072] into WMMA B-fragment order, padded to N=16.
// fcp[(kstep*32 + lane)*2 + j] = fc_w[n][k], n=lane&15, k=kstep*4+(lane>>4)*2+j
// =====================================================================
__global__ void __launch_bounds__(256) pack_fc_kernel(const float* __restrict__ fc_w,
                                                      float* __restrict__ fcp) {
  int t = blockIdx.x * blockDim.x + threadIdx.x;   // 0 .. 768*64-1
  if (t >= (INPUT_DIM / 4) * 64) return;
  int kstep = t >> 6;
  int r = t & 63;
  int lane = r >> 1;
  int j = r & 1;
  int n = lane & 15;
  int k = kstep * 4 + ((lane >> 4) << 1) + j;
  fcp[t] = (n < N_QUBITS) ? fc_w[n * INPUT_DIM + k] : 0.0f;
}

// =====================================================================
// Kernel A: feats = tanh(x @ fc_w^T + fc_b).
// Bandwidth-bound WMMA f32 GEMM; x tiles double-buffered in LDS via
// global_load_async_to_lds_b128, 8 waves split K, LDS tree-reduce + tanh.
// =====================================================================
__global__ void __launch_bounds__(256) feats_kernel(const float* __restrict__ x,
                                                    const float* __restrict__ fcp,
                                                    const float* __restrict__ fc_b,
                                                    float* __restrict__ feats) {
  __shared__ float xs[2 * TILE_F];
  __shared__ float red[8 * 256];
  const int tid  = threadIdx.x;
  const int wave = tid >> 5;
  const int lane = tid & 31;
  const int lo   = lane & 15;
  const int hi   = lane >> 4;
  const int row0 = blockIdx.x * 16;

  auto issue = [&](int blk) {
    const int kb   = blk * BK;
    const int base = (blk & 1) * TILE_F;
    // 16 rows x 128 floats = 512 float4; 256 threads -> 2 async b128 each
    #pragma unroll
    for (int j = 0; j < 2; ++j) {
      int idx = j * 256 + tid;          // float4 idx; 32 per row
      int row = idx >> 5;
      int c4  = idx & 31;
      async_ld_b128(x + (size_t)(row0 + row) * INPUT_DIM + kb + c4 * 4,
                    &xs[base + row * XS_STRIDE + c4 * 4]);
    }
  };

  v8f acc = {};
  issue(0);
  for (int blk = 0; blk < NB; ++blk) {
    wait_async0();                      // this wave's tile writes landed in LDS
    __syncthreads();                    // ... and everyone else's
    if (blk + 1 < NB) issue(blk + 1);   // DMA next tile into other buffer
    const int base = (blk & 1) * TILE_F;
    const int kb   = blk * BK;
    const int kl0  = wave * 16;         // each wave owns 16 K of this block
    #pragma unroll
    for (int s = 0; s < 4; ++s) {
      int kl = kl0 + s * 4;
      const float* ap = &xs[base + lo * XS_STRIDE + kl + hi * 2];
      v2f a; a[0] = ap[0]; a[1] = ap[1];
      int kstep = (kb + kl) >> 2;
      const float* bp = &fcp[(kstep * 32 + lane) * 2];
      v2f b; b[0] = bp[0]; b[1] = bp[1];
      acc = wmma_f32_16x16x4(a, b, acc);
    }
    // next iteration's top barrier protects the buffer being recycled
  }
  // cross-wave reduction of 16x16 C tiles
  #pragma unroll
  for (int i = 0; i < 8; ++i) {
    int m = i + hi * 8;
    red[wave * 256 + m * 16 + lo] = acc[i];
  }
  __syncthreads();
  {
    int m = tid >> 4, n = tid & 15;
    float s = 0.f;
    #pragma unroll
    for (int w = 0; w < 8; ++w) s += red[w * 256 + tid];
    if (n < N_QUBITS)
      feats[(row0 + m) * N_QUBITS + n] = tanhf(s + fc_b[n]);
  }
}

// =====================================================================
// Kernel B: simulate the weight-only circuit on 256 basis columns -> U[k][b]
// One WG (64 threads) per column; 256-amp state in LDS.
// Wire w <-> bit (7-w) of the flat index (axis 0 = MSB).
// =====================================================================
__device__ __forceinline__ void apply1(c32* st, int w, const c32 m[2][2], int tid) {
  int mask = 1 << (7 - w);
  for (int p = tid; p < 128; p += 64) {
    int i0 = ((p & ~(mask - 1)) << 1) | (p & (mask - 1));
    int i1 = i0 | mask;
    c32 a0 = st[i0], a1 = st[i1];
    st[i0] = cadd(cmul(m[0][0], a0), cmul(m[0][1], a1));
    st[i1] = cadd(cmul(m[1][0], a0), cmul(m[1][1], a1));
  }
  __syncthreads();
}

__device__ __forceinline__ void apply2(c32* st, int wa, int wb, const c32 m[4][4], int tid) {
  int ma = 1 << (7 - wa), mb = 1 << (7 - wb);
  int m1 = (ma < mb) ? ma : mb;
  int m2 = ma ^ mb ^ m1;
  int g = tid;                                   // 64 groups, 1 per thread
  int i = ((g & ~(m1 - 1)) << 1) | (g & (m1 - 1));
  i = ((i & ~(m2 - 1)) << 1) | (i & (m2 - 1));
  int idx[4];
  #pragma unroll
  for (int r = 0; r < 4; ++r) idx[r] = i | ((r >> 1) ? ma : 0) | ((r & 1) ? mb : 0);
  c32 v[4];
  #pragma unroll
  for (int r = 0; r < 4; ++r) v[r] = st[idx[r]];
  #pragma unroll
  for (int r = 0; r < 4; ++r) {
    c32 s = c32{0.f, 0.f};
    #pragma unroll
    for (int c = 0; c < 4; ++c) s = cadd(s, cmul(m[r][c], v[c]));
    st[idx[r]] = s;
  }
  __syncthreads();
}

__device__ __forceinline__ void zero4(c32 m[4][4]) {
  #pragma unroll
  for (int i = 0; i < 4; ++i)
    #pragma unroll
    for (int j = 0; j < 4; ++j) m[i][j] = c32{0.f, 0.f};
}

__device__ __forceinline__ void u3m(float t, float p, float d, c32 m[2][2]) {
  float ct, stt, cp, sp, cd, sd, cpd, spd;
  sincosf(t * 0.5f, &stt, &ct);
  sincosf(p, &sp, &cp);
  sincosf(d, &sd, &cd);
  sincosf(p + d, &spd, &cpd);
  m[0][0] = c32{ct, 0.f};
  m[0][1] = c32{-cd * stt, -sd * stt};
  m[1][0] = c32{cp * stt, sp * stt};
  m[1][1] = c32{cpd * ct, spd * ct};
}

__device__ __forceinline__ void isingZZm(float phi, c32 m[4][4]) {
  float s, c; sincosf(phi * 0.5f, &s, &c);
  zero4(m);
  m[0][0] = c32{c, -s}; m[3][3] = c32{c, -s};
  m[1][1] = c32{c,  s}; m[2][2] = c32{c,  s};
}
__device__ __forceinline__ void isingXXm(float phi, c32 m[4][4]) {
  float s, c; sincosf(phi * 0.5f, &s, &c);
  zero4(m);
  #pragma unroll
  for (int r = 0; r < 4; ++r) { m[r][r] = c32{c, 0.f}; m[r][3 - r] = c32{0.f, -s}; }
}
__device__ __forceinline__ void isingYYm(float phi, c32 m[4][4]) {
  float s, c; sincosf(phi * 0.5f, &s, &c);
  zero4(m);
  m[0][0] = c32{c,0.f}; m[1][1] = c32{c,0.f}; m[2][2] = c32{c,0.f}; m[3][3] = c32{c,0.f};
  m[0][3] = c32{0.f,  s}; m[3][0] = c32{0.f,  s};
  m[1][2] = c32{0.f, -s}; m[2][1] = c32{0.f, -s};
}
__device__ __forceinline__ void cu_mat(const c32 u[2][2], c32 m[4][4]) {
  zero4(m);
  m[0][0] = c32{1.f, 0.f}; m[1][1] = c32{1.f, 0.f};
  m[2][2] = u[0][0]; m[2][3] = u[0][1];
  m[3][2] = u[1][0]; m[3][3] = u[1][1];
}

// Pauli: 0=I 1=X 2=Y 3=Z.  P|j> = phase(j)|j^flip>
__device__ __forceinline__ c32 pphase(int p, int j) {
  if (p <= 1) return c32{1.f, 0.f};                   // I, X
  if (p == 2) return c32{0.f, j ? -1.f : 1.f};        // Y: i(1-2j)
  return c32{j ? -1.f : 1.f, 0.f};                    // Z
}
__device__ __forceinline__ int pflip(int p) { return (p == 1 || p == 2) ? 1 : 0; }

__device__ __forceinline__ void wordm(float phi, int k, c32 m[4][4]) {
  int code = k + 1;                 // skip (I,I)
  int pa = code >> 2, pb = code & 3;
  float s, c; sincosf(phi * 0.5f, &s, &c);
  zero4(m);
  int flip = (pflip(pa) << 1) | pflip(pb);
  #pragma unroll
  for (int r = 0; r < 4; ++r) {
    m[r][r] = cadd(m[r][r], c32{c, 0.f});
    int cc = r ^ flip;
    c32 ph = cmul(pphase(pa, cc >> 1), pphase(pb, cc & 1));
    c32 w  = cmul(c32{0.f, -s}, ph);  // -i*sin * W
    m[r][cc] = cadd(m[r][cc], w);
  }
}

__global__ void __launch_bounds__(64) unitary_kernel(const float* __restrict__ conv,
                                                     const float* __restrict__ pool,
                                                     const float* __restrict__ last,
                                                     c32* __restrict__ Uout) {
  __shared__ c32 st[DIM];
  const int col = blockIdx.x;
  const int tid = threadIdx.x;
  for (int i = tid; i < DIM; i += 64) st[i] = c32{(i == col) ? 1.f : 0.f, 0.f};
  __syncthreads();

  const int wiresA[8] = {0,1,2,3,4,5,6,7};
  const int wiresB[4] = {0,2,4,6};
  c32 m2[2][2]; c32 m4[4][4];

  for (int layer = 0; layer < 2; ++layer) {
    const int  nw = layer ? 4 : 8;
    const int* wl = layer ? wiresB : wiresA;
    const float* cw = conv + layer * 8 * 15;
    for (int p = 0; p < 2; ++p) {
      for (int idx = 0; idx + 1 < nw; ++idx) {
        if ((idx & 1) != p) continue;
        int wa = wl[idx], wb = wl[idx + 1];
        u3m(cw[idx*15+0],     cw[idx*15+1],     cw[idx*15+2],     m2); apply1(st, wa, m2, tid);
        u3m(cw[(idx+1)*15+3], cw[(idx+1)*15+4], cw[(idx+1)*15+5], m2); apply1(st, wb, m2, tid);
        isingZZm(cw[idx*15+6], m4); apply2(st, wa, wb, m4, tid);
        isingYYm(cw[idx*15+7], m4); apply2(st, wa, wb, m4, tid);
        isingXXm(cw[idx*15+8], m4); apply2(st, wa, wb, m4, tid);
        u3m(cw[idx*15+9],      cw[idx*15+10],     cw[idx*15+11],     m2); apply1(st, wa, m2, tid);
        u3m(cw[(idx+1)*15+12], cw[(idx+1)*15+13], cw[(idx+1)*15+14], m2); apply1(st, wb, m2, tid);
      }
    }
    const float* pw = pool + layer * 4 * 3;
    for (int idx = 1; idx < nw; idx += 2) {
      u3m(pw[(idx/2)*3+0], pw[(idx/2)*3+1], pw[(idx/2)*3+2], m2);
      cu_mat(m2, m4);
      apply2(st, wl[idx], wl[idx - 1], m4, tid);   // control = wl[idx]
    }
  }
  for (int k = 0; k < 15; ++k) { wordm(last[k], k, m4); apply2(st, 0, 4, m4, tid); }

  for (int i = tid; i < DIM; i += 64) Uout[i * DIM + col] = st[i];  // U[k][b], k-major
}

// =====================================================================
// Kernel C: M[a][b] = sum_k s_k Re(conj(U[k,a]) U[k,b]),  s_k = (k&128)?-1:+1
// Output directly in WMMA B-fragment packed order for kernel D.
// =====================================================================
__global__ void __launch_bounds__(256) mmat_kernel(const c32* __restrict__ U,
                                                   float* __restrict__ Mp) {
  __shared__ c32 tile[16 * 256];
  const int tid = threadIdx.x;          // = column b
  const int a0  = blockIdx.x * 16;
  float acc[16];
  #pragma unroll
  for (int a = 0; a < 16; ++a) acc[a] = 0.f;

  for (int kb = 0; kb < DIM; kb += 16) {
    #pragma unroll
    for (int j = 0; j < 16; ++j) tile[j * 256 + tid] = U[(kb + j) * DIM + tid];
    __syncthreads();
    #pragma unroll
    for (int kk = 0; kk < 16; ++kk) {
      float sgn = ((kb + kk) & 128) ? -1.f : 1.f;
      c32 ub = tile[kk * 256 + tid];
      float ubx = sgn * ub.x, uby = sgn * ub.y;
      #pragma unroll
      for (int a = 0; a < 16; ++a) {
        c32 ua = tile[kk * 256 + a0 + a];    // LDS broadcast
        acc[a] += ua.x * ubx + ua.y * uby;
      }
    }
    __syncthreads();
  }
  const int b = tid;
  const int nt = b >> 4, nlo = b & 15;
  #pragma unroll
  for (int a = 0; a < 16; ++a) {
    int k = a0 + a;
    int kstep = k >> 2, hi = (k >> 1) & 1, j = k & 1;
    int lane = nlo | (hi << 4);
    Mp[((nt * 64 + kstep) * 32 + lane) * 2 + j] = acc[a];
  }
}

// =====================================================================
// Kernel D: per 16 samples: psi from feats, Y = Psi * M via WMMA f32,
// qval = sum(Y .* Psi), logits = qval * out_w + out_b.
// =====================================================================
__global__ void __launch_bounds__(256) qcls_kernel(const float* __restrict__ feats,
                                                   const float* __restrict__ Mp,
                                                   const float* __restrict__ out_w,
                                                   const float* __restrict__ out_b,
                                                   float* __restrict__ out) {
  __shared__ float psi[16 * PS];
  __shared__ float cs[16][8][2];
  __shared__ float qv[16];
  const int tid  = threadIdx.x;
  const int wave = tid >> 5;
  const int lane = tid & 31;
  const int lo   = lane & 15;
  const int hi   = lane >> 4;
  const int s0   = blockIdx.x * 16;

  if (tid < 16) qv[tid] = 0.f;
  if (tid < 128) {
    int s = tid >> 3, w = tid & 7;
    float f = feats[(s0 + s) * N_QUBITS + w];
    float sn, c; sincosf(f * 0.5f, &sn, &c);
    cs[s][w][0] = c; cs[s][w][1] = sn;
  }
  __syncthreads();

  #pragma unroll
  for (int j = 0; j < 16; ++j) {
    int idx = j * 256 + tid;
    int s = idx >> 8, b = idx & 255;
    float p = 1.f;
    #pragma unroll
    for (int w = 0; w < 8; ++w) p *= cs[s][w][(b >> (7 - w)) & 1];
    psi[s * PS + b] = p;
  }
  __syncthreads();

  float priv[8];
  #pragma unroll
  for (int i = 0; i < 8; ++i) priv[i] = 0.f;

  for (int t2 = 0; t2 < 2; ++t2) {
    const int nt = wave + t2 * 8;       // each wave handles 2 N-tiles
    v8f acc = {};
    for (int kstep = 0; kstep < 64; ++kstep) {
      const float* ap = &psi[lo * PS + kstep * 4 + hi * 2];
      v2f a; a[0] = ap[0]; a[1] = ap[1];
      const float* bp = &Mp[((nt * 64 + kstep) * 32 + lane) * 2];
      v2f b; b[0] = bp[0]; b[1] = bp[1];
      acc = wmma_f32_16x16x4(a, b, acc);
    }
    const int n = nt * 16 + lo;
    #pragma unroll
    for (int i = 0; i < 8; ++i) {
      int m = i + hi * 8;
      priv[i] += acc[i] * psi[m * PS + n];
    }
  }
  // reduce over the 16 lanes of each half-wave (same m set per half)
  #pragma unroll
  for (int i = 0; i < 8; ++i) {
    float v = priv[i];
    v += __shfl_xor(v, 1, 16);
    v += __shfl_xor(v, 2, 16);
    v += __shfl_xor(v, 4, 16);
    v += __shfl_xor(v, 8, 16);
    priv[i] = v;
  }
  if (lo == 0) {
    #pragma unroll
    for (int i = 0; i < 8; ++i) atomicAdd(&qv[i + hi * 8], priv[i]);
  }
  __syncthreads();

  if (tid < 16 * NUM_CLASSES) {
    int s = tid / NUM_CLASSES, c = tid % NUM_CLASSES;
    out[(s0 + s) * NUM_CLASSES + c] = qv[s] * out_w[c] + out_b[c];
  }
}

// =====================================================================
extern "C" void kernel_launch(void* const* d_in, const int* in_sizes, int n_in,
                              void* d_out, int out_size, void* d_ws, size_t ws_size,
                              hipStream_t stream) {
  const float* x     = (const float*)d_in[0];
  const float* fc_w  = (const float*)d_in[1];
  const float* fc_b  = (const float*)d_in[2];
  const float* conv  = (const float*)d_in[3];
  const float* pool  = (const float*)d_in[4];
  const float* last  = (const float*)d_in[5];
  const float* out_w = (const float*)d_in[6];
  const float* out_b = (const float*)d_in[7];
  float* out = (float*)d_out;

  char* ws = (char*)d_ws;
  float* fcp   = (float*)(ws);                               // 768*64*4      = 196608 B
  c32*   U     = (c32*)  (ws + 196608);                      // 256*256*8     = 524288 B
  float* Mp    = (float*)(ws + 196608 + 524288);             // 256*256*4     = 262144 B
  float* feats = (float*)(ws + 196608 + 524288 + 262144);    // 8192*8*4      = 262144 B

  pack_fc_kernel<<<192, 256, 0, stream>>>(fc_w, fcp);
  unitary_kernel<<<256, 64, 0, stream>>>(conv, pool, last, U);
  feats_kernel<<<512, 256, 0, stream>>>(x, fcp, fc_b, feats);
  mmat_kernel<<<16, 256, 0, stream>>>(U, Mp);
  qcls_kernel<<<512, 256, 0, stream>>>(feats, Mp, out_w, out_b, out);
}